// NS_V2_47064251629926
// MI455X (gfx1250) — compile-verified
//
#include <hip/hip_runtime.h>
#include <stdint.h>

// Problem constants (fixed by the reference).
#define NCLS    8
#define BATCH   16
#define HDIM    512
#define WDIM    512
#define HW      (HDIM * WDIM)          // 262144
#define EPS_F   1e-12f

// Pass-2 tiling: 128x8 pixels per 256-thread block (8 wave32 waves),
// 4 consecutive pixels per thread -> all VMEM is b128.
#define TILE_W   128
#define TILE_H   8
#define HALO_W   (TILE_W + 2)          // 130 (logical)
#define HALO_WP  132                   // padded stride, multiple of 4
#define HALO_H   (TILE_H + 2)          // 10
#define HALO_N   (HALO_H * HALO_W)     // 1300 bytes staged

// ---------------------------------------------------------------------------
// Pass 1: per-pixel argmax over 8 class planes. 4 pixels/thread:
// 8x global_load_b128, pack 4 class ids into one u32 store.
// ---------------------------------------------------------------------------
__global__ void __launch_bounds__(256)
ns_argmax_kernel(const float* __restrict__ logits, uint32_t* __restrict__ yhard4) {
    int p4 = blockIdx.x * 256 + threadIdx.x;           // over BATCH*HW/4
    if (p4 >= BATCH * (HW / 4)) return;
    int b   = p4 >> 16;                                // / (HW/4) = 65536
    int hw4 = p4 & 0xFFFF;

    const float4* base4 = (const float4*)(logits + (size_t)b * NCLS * HW) + hw4;

    float4 v0 = base4[0];
    const float* vf0 = reinterpret_cast<const float*>(&v0);
    float bv[4];
    int   bi[4];
#pragma unroll
    for (int k = 0; k < 4; ++k) { bv[k] = vf0[k]; bi[k] = 0; }

#pragma unroll
    for (int c = 1; c < NCLS; ++c) {
        float4 v = base4[(size_t)c * (HW / 4)];
        const float* vf = reinterpret_cast<const float*>(&v);
#pragma unroll
        for (int k = 0; k < 4; ++k) {
            if (vf[k] > bv[k]) { bv[k] = vf[k]; bi[k] = c; }  // strict > == jnp.argmax
        }
    }
    uint32_t pack = (uint32_t)bi[0] | ((uint32_t)bi[1] << 8) |
                    ((uint32_t)bi[2] << 16) | ((uint32_t)bi[3] << 24);
    yhard4[p4] = pack;
}

// ---------------------------------------------------------------------------
// Pass 2: async-DMA the class-map halo tile into LDS, 3x3 agreement count,
// temperature gather, then b128 scale of all 8 class planes.
// ---------------------------------------------------------------------------
__global__ void __launch_bounds__(256)
ns_temp_scale_kernel(const float* __restrict__ logits,
                     const float* __restrict__ nt,         // [8,9]
                     const uint8_t* __restrict__ yhard,
                     float* __restrict__ out) {
    __shared__ uint32_t s_y32[(HALO_H * HALO_WP) / 4];     // 1320 B, dword-aligned
    __shared__ float    s_nt[NCLS * 9];
    int8_t* s_y = (int8_t*)s_y32;

    const int TX = WDIM / TILE_W;                          // 4
    const int TY = HDIM / TILE_H;                          // 64

    int blk = blockIdx.x;
    int b   = blk >> 8;                                    // / (TX*TY) = 256
    int t   = blk & 0xFF;
    int tyb = t >> 2;                                      // / TX
    int txb = t & 3;
    int h0  = tyb * TILE_H;
    int w0  = txb * TILE_W;

    int tid = threadIdx.x;
    if (tid < NCLS * 9) s_nt[tid] = nt[tid];

    // --- async halo stage: 1300 bytes, clamped addresses (always in-bounds) ---
    const uint8_t* ybase = yhard + (size_t)b * HW;
    for (int e = tid; e < HALO_N; e += 256) {
        int r  = e / HALO_W;
        int c  = e - r * HALO_W;
        int gh = min(max(h0 + r - 1, 0), HDIM - 1);
        int gw = min(max(w0 + c - 1, 0), WDIM - 1);
        uint64_t gaddr = (uint64_t)(const void*)(ybase + gh * WDIM + gw);
        uint32_t laddr = (uint32_t)(uintptr_t)(s_y + r * HALO_WP + c);
        // CDNA5 async DMA: copies one byte from the global address in gaddr to the LDS address in laddr, tracked by ASYNCcnt.
        asm volatile("global_load_async_to_lds_b8 %0, %1, off"
                     :: "v"(laddr), "v"(gaddr)
                     : "memory");
    }
    asm volatile("s_wait_asynccnt 0" ::: "memory");
    __syncthreads();

    // --- per thread: 4 consecutive pixels in one row ---
    int lty   = tid >> 5;                                  // 0..7
    int ltx   = tid & 31;                                  // 0..31
    int h     = h0 + lty;
    int wbase = w0 + (ltx << 2);

    // Read the 3 stencil rows as two aligned dwords each (6-byte window used).
    uint8_t rb[3][6];
#pragma unroll
    for (int di = 0; di < 3; ++di) {
        int off = (lty + di) * HALO_WP + (ltx << 2);       // dword-aligned
        uint32_t lo = *(const uint32_t*)&s_y[off];
        uint32_t hi = *(const uint32_t*)&s_y[off + 4];
#pragma unroll
        for (int j = 0; j < 4; ++j) rb[di][j] = (uint8_t)(lo >> (8 * j));
        rb[di][4] = (uint8_t)hi;
        rb[di][5] = (uint8_t)(hi >> 8);
    }

    bool rv[3], cv[6];
#pragma unroll
    for (int di = 0; di < 3; ++di)
        rv[di] = (unsigned)(h + di - 1) < (unsigned)HDIM;
#pragma unroll
    for (int j = 0; j < 6; ++j)
        cv[j] = (unsigned)(wbase + j - 1) < (unsigned)WDIM;

    float inv[4];
#pragma unroll
    for (int k = 0; k < 4; ++k) {
        int center = rb[1][k + 1];
        int cnt = -1;                                      // cancels the self-match
#pragma unroll
        for (int di = 0; di < 3; ++di) {
#pragma unroll
            for (int dj = 0; dj < 3; ++dj) {
                cnt += (rv[di] && cv[k + dj] && (rb[di][k + dj] == center)) ? 1 : 0;
            }
        }
        float tv = s_nt[center * 9 + cnt];
        tv = fmaxf(tv, 0.0f) + EPS_F;
        inv[k] = 1.0f / tv;
    }

    // --- b128 scale & write of all 8 class planes ---
    size_t elem = (size_t)b * NCLS * HW + (size_t)h * WDIM + wbase;
    const float4* src4 = (const float4*)(logits + elem);
    float4*       dst4 = (float4*)(out + elem);
#pragma unroll
    for (int c = 0; c < NCLS; ++c) {
        float4 v = src4[(size_t)c * (HW / 4)];
        float* vf = reinterpret_cast<float*>(&v);
#pragma unroll
        for (int k = 0; k < 4; ++k) vf[k] *= inv[k];
        dst4[(size_t)c * (HW / 4)] = v;
    }
}

// ---------------------------------------------------------------------------
extern "C" void kernel_launch(void* const* d_in, const int* in_sizes, int n_in,
                              void* d_out, int out_size, void* d_ws, size_t ws_size,
                              hipStream_t stream) {
    const float* logits = (const float*)d_in[0];           // [16,8,512,512] f32
    const float* nt     = (const float*)d_in[1];           // [8,9] f32
    float*       out    = (float*)d_out;                   // [16,8,512,512] f32
    uint8_t*     yhard  = (uint8_t*)d_ws;                  // 4 MiB class map scratch

    const int npix4 = BATCH * (HW / 4);                    // 1,048,576
    ns_argmax_kernel<<<npix4 / 256, 256, 0, stream>>>(logits, (uint32_t*)yhard);

    const int nblk = BATCH * (HDIM / TILE_H) * (WDIM / TILE_W);  // 4096
    ns_temp_scale_kernel<<<nblk, 256, 0, stream>>>(logits, nt, yhard, out);
}